// FraudDetector_80290118632119
// MI455X (gfx1250) — compile-verified
//
#include <hip/hip_runtime.h>
#include <cstddef>

typedef float v2f __attribute__((ext_vector_type(2)));
typedef float v8f __attribute__((ext_vector_type(8)));

// ---------- ordered-int encoding for float atomic max ----------
__device__ __forceinline__ unsigned fkey(float f) {
  unsigned b = __float_as_uint(f);
  return (b & 0x80000000u) ? ~b : (b | 0x80000000u);
}
__device__ __forceinline__ float fdec(unsigned k) {
  unsigned b = (k & 0x80000000u) ? (k & 0x7FFFFFFFu) : ~k;
  return __uint_as_float(b);
}

// ---------- zero fill (graph-safe init) ----------
__global__ void fill_zero_kernel(float* __restrict__ p, size_t n) {
  size_t t = (size_t)blockIdx.x * blockDim.x + threadIdx.x;
  if (t < n) p[t] = 0.0f;
}

// ---------- C[N,OUT] = X[N,K] @ W[OUT,K]^T (+bias), fp32 WMMA ----------
// One wave computes a full 16 x OUT strip: A fragment loaded ONCE per k-step,
// reused across all NCOLT column tiles (X read exactly once from HBM).
// V_WMMA_F32_16X16X4_F32 layouts:
//  A(16x4): lanes 0-15 -> M=lane (K=k0,k0+1); lanes 16-31 -> M=lane-16 (K=k0+2,k0+3)
//  B(4x16): lane = column N, same K split.  C: vgpr j -> M=j+8*(lane>=16), N=lane&15
template <int NCOLT>
__global__ void gemm_xwT_kernel(const float* __restrict__ X, const float* __restrict__ W,
                                const float* __restrict__ bias, float* __restrict__ C,
                                int N, int K) {
  const int OUT = NCOLT * 16;
  const int lane = threadIdx.x & 31;
  const int wave = threadIdx.x >> 5;
  const int nRowTiles = N >> 4;                 // N is a multiple of 16
  const int rowTile = blockIdx.x * 8 + wave;
  if (rowTile >= nRowTiles) return;             // wave-uniform skip
  const int half = lane >> 4;                   // 0 or 1
  const int l16  = lane & 15;
  const int row = rowTile * 16 + l16;
  const float* xrow  = X + (size_t)row * K + half * 2;
  const float* wbase = W + (size_t)l16 * K + half * 2;

  // run-ahead prefetch of this lane's X row (lowers to global_prefetch_b8)
  __builtin_prefetch(xrow, 0, 1);
  if (K > 32)  __builtin_prefetch(xrow + 32, 0, 1);
  if (K > 64)  __builtin_prefetch(xrow + 64, 0, 1);
  if (K > 96)  __builtin_prefetch(xrow + 96, 0, 1);

  const v8f vzero = {0.f, 0.f, 0.f, 0.f, 0.f, 0.f, 0.f, 0.f};
  v8f acc[NCOLT];
#pragma unroll
  for (int ct = 0; ct < NCOLT; ++ct) acc[ct] = vzero;

  for (int k0 = 0; k0 < K; k0 += 4) {
    v2f a = *reinterpret_cast<const v2f*>(xrow + k0);
#pragma unroll
    for (int ct = 0; ct < NCOLT; ++ct) {
      v2f b = *reinterpret_cast<const v2f*>(wbase + (size_t)ct * 16 * K + k0);
      acc[ct] = __builtin_amdgcn_wmma_f32_16x16x4_f32(false, a, false, b, (short)0,
                                                      acc[ct], false, false);
    }
  }

#pragma unroll
  for (int ct = 0; ct < NCOLT; ++ct) {
    const int col = ct * 16 + l16;
    const float bv = bias ? bias[col] : 0.0f;
#pragma unroll
    for (int j = 0; j < 8; ++j) {
      int r = rowTile * 16 + j + half * 8;
      C[(size_t)r * OUT + col] = acc[ct][j] + bv;
    }
  }
}

// ---------- fold attention vectors: v[h,k] = sum_c a[h,c] * w[h*64+c, k] ----------
// vout[256]: [0..127] = v_src (h0 then h1), [128..255] = v_dst
__global__ void fold_att_kernel(const float* __restrict__ aS, const float* __restrict__ wS,
                                const float* __restrict__ aD, const float* __restrict__ wD,
                                float* __restrict__ vout) {
  int t = threadIdx.x;            // 0..255
  int which = t >> 7;             // 0 = src, 1 = dst
  int h = (t >> 6) & 1;
  int k = t & 63;
  const float* a = which ? aD : aS;
  const float* w = which ? wD : wS;
  float s = 0.0f;
  for (int c = 0; c < 64; ++c) s += a[h * 64 + c] * w[(h * 64 + c) * 64 + k];
  vout[t] = s;
}

// ---------- al[n,h] = x[n,:64] . v[h,:64], wave-per-node ----------
__global__ void attn_scalar_kernel(const float* __restrict__ x, const float* __restrict__ v,
                                   float* __restrict__ al, int N) {
  int lane = threadIdx.x & 31;
  int n = blockIdx.x * 8 + (threadIdx.x >> 5);
  if (n >= N) return;
  const float* xr = x + (size_t)n * 64;
  float x0 = xr[lane], x1 = xr[lane + 32];
  float p0 = x0 * v[lane]      + x1 * v[lane + 32];
  float p1 = x0 * v[64 + lane] + x1 * v[96 + lane];
#pragma unroll
  for (int off = 16; off > 0; off >>= 1) {
    p0 += __shfl_xor(p0, off, 32);
    p1 += __shfl_xor(p1, off, 32);
  }
  if (lane == 0) { al[(size_t)n * 2] = p0; al[(size_t)n * 2 + 1] = p1; }
}

// ---------- edge pass 1: e = lrelu(al_s[src]+al_d[dst]); segment max ----------
__global__ void edge_max_kernel(const int* __restrict__ src, const int* __restrict__ dst,
                                const float* __restrict__ al_s, const float* __restrict__ al_d,
                                float* __restrict__ ebuf, unsigned* __restrict__ mkey, int E) {
  int t = blockIdx.x * blockDim.x + threadIdx.x;
  if (t >= E * 2) return;
  int e = t >> 1, h = t & 1;
  int si = src[e], di = dst[e];
  float v = al_s[(size_t)si * 2 + h] + al_d[(size_t)di * 2 + h];
  v = v > 0.0f ? v : 0.2f * v;
  ebuf[t] = v;
  atomicMax(&mkey[(size_t)di * 2 + h], fkey(v));
}

// ---------- edge pass 2: p = exp(e - m[dst]); segment sum ----------
__global__ void edge_sum_kernel(const int* __restrict__ dst, const float* __restrict__ ebuf,
                                const unsigned* __restrict__ mkey, float* __restrict__ pbuf,
                                float* __restrict__ ssum, int E) {
  int t = blockIdx.x * blockDim.x + threadIdx.x;
  if (t >= E * 2) return;
  int e = t >> 1, h = t & 1;
  int di = dst[e];
  float m = fdec(mkey[(size_t)di * 2 + h]);
  float p = expf(ebuf[t] - m);
  pbuf[t] = p;
  atomicAdd(&ssum[(size_t)di * 2 + h], p);
}

// ---------- edge pass 3: acc[dst] += alpha * hs[src], wave-per-edge ----------
__global__ void edge_scatter_kernel(const int* __restrict__ src, const int* __restrict__ dst,
                                    const float* __restrict__ hs, const float* __restrict__ pbuf,
                                    const float* __restrict__ ssum, float* __restrict__ acc, int E) {
  int lane = threadIdx.x & 31;
  int e = (blockIdx.x * blockDim.x + threadIdx.x) >> 5;
  if (e >= E) return;
  int si = src[e], di = dst[e];
  int ch = lane * 4;              // 4 channels per lane, 128 total
  int h = ch >> 6;
  float alpha = pbuf[(size_t)e * 2 + h] / (ssum[(size_t)di * 2 + h] + 1e-16f);
  const float4 hv = *reinterpret_cast<const float4*>(hs + (size_t)si * 128 + ch);
  float* ap = acc + (size_t)di * 128 + ch;
  atomicAdd(ap + 0, hv.x * alpha);
  atomicAdd(ap + 1, hv.y * alpha);
  atomicAdd(ap + 2, hv.z * alpha);
  atomicAdd(ap + 3, hv.w * alpha);
}

// ---------- finalize: h = relu(mean_heads(acc) + sum of relation biases) ----------
__global__ void finalize_kernel(const float* __restrict__ acc, const float* __restrict__ b0,
                                const float* __restrict__ b1, const float* __restrict__ b2,
                                float* __restrict__ hout, int N) {
  int t = blockIdx.x * blockDim.x + threadIdx.x;
  if (t >= N * 64) return;
  int n = t >> 6, c = t & 63;
  float v = (acc[(size_t)n * 128 + c] + acc[(size_t)n * 128 + 64 + c]) * 0.5f + b0[c];
  if (b1) v += b1[c];
  if (b2) v += b2[c];
  hout[t] = v > 0.0f ? v : 0.0f;
}

// ---------- classifier: out = relu(h @ w1.T + b1) @ w2.T + b2, wave-per-node ----------
__global__ void classifier_kernel(const float* __restrict__ h, const float* __restrict__ w1,
                                  const float* __restrict__ b1, const float* __restrict__ w2,
                                  const float* __restrict__ b2, float* __restrict__ out, int N) {
  int lane = threadIdx.x & 31;
  int n = (blockIdx.x * blockDim.x + threadIdx.x) >> 5;
  if (n >= N) return;
  const float* hr = h + (size_t)n * 64;
  const float* wr = w1 + lane * 64;
  float t = b1[lane];
  for (int c = 0; c < 64; ++c) t += hr[c] * wr[c];
  t = t > 0.0f ? t : 0.0f;
  float u = t * w2[lane];
#pragma unroll
  for (int off = 16; off > 0; off >>= 1) u += __shfl_xor(u, off, 32);
  if (lane == 0) out[n] = u + b2[0];
}

extern "C" void kernel_launch(void* const* d_in, const int* in_sizes, int n_in,
                              void* d_out, int out_size, void* d_ws, size_t ws_size,
                              hipStream_t stream) {
  const float* x_post = (const float*)d_in[0];
  const float* x_user = (const float*)d_in[1];
  const float* x_ent  = (const float*)d_in[2];
  const int* src_e[6]; const int* dst_e[6]; int E_e[6];
  for (int r = 0; r < 6; ++r) {
    src_e[r] = (const int*)d_in[3 + 2 * r];
    dst_e[r] = (const int*)d_in[4 + 2 * r];
    E_e[r]   = in_sizes[3 + 2 * r];
  }
  const float* post_w = (const float*)d_in[15];
  const float* post_b = (const float*)d_in[16];
  const float* user_w = (const float*)d_in[17];
  const float* user_b = (const float*)d_in[18];
  const float* ent_w  = (const float*)d_in[19];
  const float* ent_b  = (const float*)d_in[20];
  const float* Wsrc   = (const float*)d_in[21];  // [2][6][128][64]
  const float* Wdst   = (const float*)d_in[22];
  const float* Asrc   = (const float*)d_in[23];  // [2][6][2][64]
  const float* Adst   = (const float*)d_in[24];
  const float* Gb     = (const float*)d_in[25];  // [2][6][64]
  const float* c_w1   = (const float*)d_in[26];
  const float* c_b1   = (const float*)d_in[27];
  const float* c_w2   = (const float*)d_in[28];
  const float* c_b2   = (const float*)d_in[29];

  const int NP = in_sizes[0] / 768;
  const int NU = in_sizes[1] / 32;
  const int NEn = in_sizes[2] / 64;
  int maxE = 0;
  for (int r = 0; r < 6; ++r) if (E_e[r] > maxE) maxE = E_e[r];

  // ---- carve workspace ----
  char* wsp = (char*)d_ws;
  auto carve = [&](size_t bytes) -> float* {
    float* p = (float*)wsp;
    wsp += (bytes + 255) & ~(size_t)255;
    return p;
  };
  float* hP[2] = { carve((size_t)NP * 64 * 4), carve((size_t)NP * 64 * 4) };
  float* hU[2] = { carve((size_t)NU * 64 * 4), carve((size_t)NU * 64 * 4) };
  float* hE[2] = { carve((size_t)NEn * 64 * 4), carve((size_t)NEn * 64 * 4) };
  float* hs_buf   = carve((size_t)NP * 128 * 4);
  float* acc_post = carve((size_t)NP * 128 * 4);
  float* acc_user = carve((size_t)NU * 128 * 4);
  float* acc_ent  = carve((size_t)NEn * 128 * 4);
  float* al_s = carve((size_t)NP * 2 * 4);
  float* al_d = carve((size_t)NP * 2 * 4);
  float* mkey = carve((size_t)NP * 2 * 4);
  float* ssum = carve((size_t)NP * 2 * 4);
  float* ebuf = carve((size_t)maxE * 2 * 4);
  float* vbuf = carve(256 * 4);

  auto gemm = [&](const float* X, const float* W, const float* bias, float* C,
                  int N, int K, int OUT) {
    const unsigned gx = (unsigned)((N / 16 + 7) / 8);
    if (OUT == 64)
      gemm_xwT_kernel<4><<<gx, 256, 0, stream>>>(X, W, bias, C, N, K);
    else
      gemm_xwT_kernel<8><<<gx, 256, 0, stream>>>(X, W, bias, C, N, K);
  };
  auto zero = [&](float* p, size_t n) {
    fill_zero_kernel<<<(unsigned)((n + 255) / 256), 256, 0, stream>>>(p, n);
  };

  // ---- input projections (fp32 WMMA) ----
  gemm(x_post, post_w, post_b, hP[0], NP, 768, 64);
  gemm(x_user, user_w, user_b, hU[0], NU, 32, 64);
  gemm(x_ent,  ent_w,  ent_b,  hE[0], NEn, 64, 64);

  int cur = 0;
  for (int l = 0; l < 2; ++l) {
    zero(acc_post, (size_t)NP * 128);
    zero(acc_user, (size_t)NU * 128);
    zero(acc_ent,  (size_t)NEn * 128);

    // relations: 0 publish U->P, 1 repost U->P, 2 contain P->E,
    //            3 interact U->U, 4 follow U->U, 5 similar P->P
    const float* xsA[6] = { hU[cur], hU[cur], hP[cur], hU[cur], hU[cur], hP[cur] };
    const int    nsA[6] = { NU, NU, NP, NU, NU, NP };
    const float* xdA[6] = { hP[cur], hP[cur], hE[cur], hU[cur], hU[cur], hP[cur] };
    const int    ndA[6] = { NP, NP, NEn, NU, NU, NP };
    float*      accA[6] = { acc_post, acc_post, acc_ent, acc_user, acc_user, acc_post };

    for (int r = 0; r < 6; ++r) {
      const float* wS = Wsrc + (size_t)(l * 6 + r) * 128 * 64;
      const float* wD = Wdst + (size_t)(l * 6 + r) * 128 * 64;
      const float* aS = Asrc + (size_t)(l * 6 + r) * 2 * 64;
      const float* aD = Adst + (size_t)(l * 6 + r) * 2 * 64;
      const int ns = nsA[r], nd = ndA[r], E = E_e[r];

      fold_att_kernel<<<1, 256, 0, stream>>>(aS, wS, aD, wD, vbuf);
      gemm(xsA[r], wS, nullptr, hs_buf, ns, 64, 128);
      attn_scalar_kernel<<<(unsigned)((ns + 7) / 8), 256, 0, stream>>>(xsA[r], vbuf, al_s, ns);
      attn_scalar_kernel<<<(unsigned)((nd + 7) / 8), 256, 0, stream>>>(xdA[r], vbuf + 128, al_d, nd);
      zero(mkey, (size_t)nd * 2);
      zero(ssum, (size_t)nd * 2);

      const unsigned gE2 = (unsigned)((E * 2 + 255) / 256);
      edge_max_kernel<<<gE2, 256, 0, stream>>>(src_e[r], dst_e[r], al_s, al_d,
                                               ebuf, (unsigned*)mkey, E);
      edge_sum_kernel<<<gE2, 256, 0, stream>>>(dst_e[r], ebuf, (const unsigned*)mkey,
                                               ebuf, ssum, E);
      const unsigned gE32 = (unsigned)(((long long)E * 32 + 255) / 256);
      edge_scatter_kernel<<<gE32, 256, 0, stream>>>(src_e[r], dst_e[r], hs_buf,
                                                    ebuf, ssum, accA[r], E);
    }

    const int nxt = 1 - cur;
    finalize_kernel<<<(unsigned)(((size_t)NP * 64 + 255) / 256), 256, 0, stream>>>(
        acc_post, Gb + (size_t)(l * 6 + 0) * 64, Gb + (size_t)(l * 6 + 1) * 64,
        Gb + (size_t)(l * 6 + 5) * 64, hP[nxt], NP);
    finalize_kernel<<<(unsigned)(((size_t)NU * 64 + 255) / 256), 256, 0, stream>>>(
        acc_user, Gb + (size_t)(l * 6 + 3) * 64, Gb + (size_t)(l * 6 + 4) * 64,
        nullptr, hU[nxt], NU);
    finalize_kernel<<<(unsigned)(((size_t)NEn * 64 + 255) / 256), 256, 0, stream>>>(
        acc_ent, Gb + (size_t)(l * 6 + 2) * 64, nullptr, nullptr, hE[nxt], NEn);
    cur = nxt;
  }

  classifier_kernel<<<(unsigned)(((size_t)NP * 32 + 255) / 256), 256, 0, stream>>>(
      hP[cur], c_w1, c_b1, c_w2, c_b2, (float*)d_out, NP);
}